// StadvFlow_27797028340216
// MI455X (gfx1250) — compile-verified
//
#include <hip/hip_runtime.h>

// Bilinear flow warp: out[b,c,y,x] = bilinear_sample(images[b,c], y+flow[b,0], x+flow[b,1])
// Shapes fixed by the reference: B=32, C=3, H=W=512, float32.

typedef float v4f __attribute__((ext_vector_type(4)));  // true clang vector, 16B aligned

constexpr int B_ = 32;
constexpr int C_ = 3;
constexpr int H_ = 512;
constexpr int W_ = 512;
constexpr int HW = H_ * W_;            // 262144
constexpr int PPT = 4;                 // pixels per thread (b128 along W)
constexpr int GROUPS_PER_IMG = HW / PPT;   // 65536
constexpr int THREADS = 256;
constexpr int TOTAL_GROUPS = B_ * GROUPS_PER_IMG;  // 2,097,152
constexpr int BLOCKS = TOTAL_GROUPS / THREADS;     // 8192

__global__ __launch_bounds__(THREADS) void StadvFlow_warp_kernel(
    const float* __restrict__ images,   // [B,3,H,W]
    const float* __restrict__ flows,    // [B,2,H,W]  ch0 -> dy, ch1 -> dx
    float* __restrict__ out)            // [B,3,H,W]
{
    const int gid = blockIdx.x * THREADS + threadIdx.x;

    // GROUPS_PER_IMG == 65536, W_/PPT == 128 -> cheap bit ops
    const int b   = gid >> 16;          // gid / 65536
    const int rem = gid & 0xFFFF;
    const int y   = rem >> 7;           // rem / 128
    const int x   = (rem & 127) << 2;   // 4-pixel group start, 16B aligned

    const size_t flowBase = (size_t)b * 2 * HW + (size_t)y * W_ + x;
    // Flows are read exactly once: non-temporal loads keep them out of L2
    // so the 100 MB image set stays resident (L2 = 192 MB).
    const v4f fy4 = __builtin_nontemporal_load(
        reinterpret_cast<const v4f*>(flows + flowBase));          // dy
    const v4f fx4 = __builtin_nontemporal_load(
        reinterpret_cast<const v4f*>(flows + flowBase + HW));     // dx

    // Per-pixel coordinates and bilinear weights, computed once, reused for 3 channels.
    int   base[PPT];                    // y0*W + x0
    float wx0[PPT], wx1[PPT], wy0[PPT], wy1[PPT];
    const float yf = (float)y;

    #pragma unroll
    for (int j = 0; j < PPT; ++j) {
        float sx = fminf(fmaxf((float)(x + j) + fx4[j], 0.0f), (float)(W_ - 1));
        float sy = fminf(fmaxf(yf        + fy4[j], 0.0f), (float)(H_ - 1));
        int x0 = min((int)sx, W_ - 2);  // sx >= 0 -> trunc == floor
        int y0 = min((int)sy, H_ - 2);
        wx1[j] = sx - (float)x0;
        wx0[j] = (float)(x0 + 1) - sx;
        wy1[j] = sy - (float)y0;
        wy0[j] = (float)(y0 + 1) - sy;
        base[j] = y0 * W_ + x0;
    }

    const float* imgB  = images + (size_t)b * C_ * HW;
    float*       outB  = out    + (size_t)b * C_ * HW + (size_t)y * W_ + x;

    #pragma unroll
    for (int c = 0; c < C_; ++c) {
        const float* img = imgB + (size_t)c * HW;
        v4f r;
        #pragma unroll
        for (int j = 0; j < PPT; ++j) {
            const float* p = img + base[j];
            // 4 taps: two adjacent pairs (x0, x0+1) on rows y0 and y0+1.
            float Ia = p[0];
            float Ic = p[1];
            float Ib = p[W_];
            float Id = p[W_ + 1];
            float top = fmaf(wx0[j], Ia, wx1[j] * Ic);
            float bot = fmaf(wx0[j], Ib, wx1[j] * Id);
            r[j] = fmaf(wy0[j], top, wy1[j] * bot);
        }
        // Output written once: non-temporal store, don't pollute L2.
        __builtin_nontemporal_store(r, reinterpret_cast<v4f*>(outB + (size_t)c * HW));
    }
}

extern "C" void kernel_launch(void* const* d_in, const int* in_sizes, int n_in,
                              void* d_out, int out_size, void* d_ws, size_t ws_size,
                              hipStream_t stream) {
    (void)in_sizes; (void)n_in; (void)out_size; (void)d_ws; (void)ws_size;
    const float* images = (const float*)d_in[0];
    const float* flows  = (const float*)d_in[1];
    float* out          = (float*)d_out;
    StadvFlow_warp_kernel<<<BLOCKS, THREADS, 0, stream>>>(images, flows, out);
}